// AGNNConv_3178275799598
// MI455X (gfx1250) — compile-verified
//
#include <hip/hip_runtime.h>
#include <hip/hip_bf16.h>
#include <math.h>

// AGNN conv for MI455X (gfx1250, wave32).
// Wave-per-row segment softmax + SPMM. x (25.6MB) is L2-resident -> the
// kernel is gather-bound from L2. Dots run on the matrix pipe via
// V_WMMA_F32_16X16X4_F32 (full fp32). Pass 1 computes online-softmax stats
// and stashes raw sims in LDS (cap 512/row); a lane-parallel sweep converts
// sims -> probabilities in place (one exp per EDGE, not per lane); pass 2 is
// then a pure SPMM stream. Wave-uniform WMMA-recompute fallback for
// degree > 512 keeps arbitrary inputs correct.

#define N_NODES 100000
#define D_FEAT  64
#define EPS_F   1e-7f
#define SIM_CAP 512            // per-row LDS sim capacity (2KB/wave)

typedef __attribute__((ext_vector_type(2))) float v2f;
typedef __attribute__((ext_vector_type(8))) float v8f;

__device__ __forceinline__ int lower_bound_i32(const int* __restrict__ a, int n, int v) {
    int lo = 0, hi = n;
    while (lo < hi) {
        int mid = (lo + hi) >> 1;
        if (a[mid] < v) lo = mid + 1; else hi = mid;
    }
    return lo;
}

// Kernel A: per-node L2 norms. One wave per node, float2 per lane (256B/row,
// fully coalesced), cross-lane reduce with shfl_xor (wave32).
__global__ __launch_bounds__(256) void node_norm_kernel(
        const float* __restrict__ x, float* __restrict__ norms) {
    const int node = (blockIdx.x * blockDim.x + threadIdx.x) >> 5;
    const int lane = threadIdx.x & 31;
    if (node >= N_NODES) return;
    const float2 v = *(const float2*)(x + (size_t)node * D_FEAT + 2 * lane);
    float ss = v.x * v.x + v.y * v.y;
#pragma unroll
    for (int off = 16; off; off >>= 1) ss += __shfl_xor(ss, off, 32);
    if (lane == 0) norms[node] = sqrtf(ss);
}

__global__ __launch_bounds__(256) void agnn_row_kernel(
        const float* __restrict__ x, const float* __restrict__ beta,
        const int* __restrict__ erow, const int* __restrict__ ecol,
        const float* __restrict__ norms, float* __restrict__ out, int n_edges) {
    __shared__ float s_sims[8][SIM_CAP];             // 16KB / block (8 waves)

    const int row  = (blockIdx.x * blockDim.x + threadIdx.x) >> 5;
    const int lane = threadIdx.x & 31;
    const int wib  = threadIdx.x >> 5;               // wave-in-block
    if (row >= N_NODES) return;                      // wave-uniform (wave32)

    const int start = lower_bound_i32(erow, n_edges, row);
    const int end   = start + lower_bound_i32(erow + start, n_edges - start, row + 1);
    const int deg   = end - start;

    float* orow = out + (size_t)row * D_FEAT;
    if (deg == 0) {                                  // wave-uniform
        *(float2*)(orow + 2 * lane) = make_float2(0.f, 0.f);
        return;
    }

    const float bscale = beta[0];
    const float ns     = norms[row];
    // WMMA f32 16x16x4 A-layout: lane L holds M = L&15, K = 2*(L>>4)+{0,1}.
    const int   koff   = (lane >> 4) << 1;           // 0 (lanes 0-15) or 2 (16-31)
    const int   mA     = lane & 15;                  // A-matrix row this lane feeds
    const int   mBase  = (lane < 16) ? 0 : 8;        // C-layout: lane holds M = mBase+i
    const float* __restrict__ xrow = x + (size_t)row * D_FEAT;
    const float ninf = -__builtin_inff();

    float runm = ninf, runs = 0.f;
    const int ngroups = (deg + 15) >> 4;

    // ---------- pass 1: streaming softmax stats; stash sims in LDS ----------
    for (int t = 0; t < ngroups; ++t) {
        const int e  = start + t * 16 + mA;
        const int ec = ecol[(e < end) ? e : start];  // clamp -> safe address
        const float nc = norms[ec];
        const float* __restrict__ xb = x + (size_t)ec * D_FEAT + koff;

        // D[m,n] = sum_f x[col_m][f] * xrow[f]  (replicated over n)
        v8f acc = {};
#pragma unroll
        for (int c = 0; c < 16; ++c) {
            const float2 a2 = *(const float2*)(xb + 4 * c);
            const float2 b2 = *(const float2*)(xrow + koff + 4 * c);
            v2f a; a.x = a2.x; a.y = a2.y;
            v2f b; b.x = b2.x; b.y = b2.y;
            acc = __builtin_amdgcn_wmma_f32_16x16x4_f32(
                      false, a, false, b, (short)0, acc, false, false);
        }

        float gmax = ninf;
        float sims[8];
#pragma unroll
        for (int i = 0; i < 8; ++i) {
            const int em = t * 16 + mBase + i;                 // edge-in-segment
            const float ncb = __shfl(nc, mBase + i, 32);       // norm of col_m
            // 1-ulp v_rcp_f32 instead of the ~7-op IEEE divide chain; softmax
            // is insensitive to ~1e-7 relative error in the logit.
            const float s = bscale * acc[i] * __builtin_amdgcn_rcpf(ns * ncb + EPS_F);
            sims[i] = (em < deg) ? s : ninf;
            gmax = fmaxf(gmax, sims[i]);
        }
        // Stash this group's sims (one writer per C half: lanes 0 and 16).
        if (t * 16 < SIM_CAP && (lane & 15) == 0) {
            float* dst = &s_sims[wib][t * 16 + mBase];
            *(float4*)(dst + 0) = make_float4(sims[0], sims[1], sims[2], sims[3]);
            *(float4*)(dst + 4) = make_float4(sims[4], sims[5], sims[6], sims[7]);
        }
        gmax = fmaxf(gmax, __shfl_xor(gmax, 16, 32));          // both C halves
        const float newm = fmaxf(runm, gmax);                  // finite (>=1 valid)
        float lsum = 0.f;
#pragma unroll
        for (int i = 0; i < 8; ++i) lsum += expf(sims[i] - newm);
        const float gsum = lsum + __shfl_xor(lsum, 16, 32);
        runs = runs * expf(runm - newm) + gsum;
        runm = newm;
    }

    // One precise division per ROW; everything downstream multiplies.
    const float rinv = 1.0f / runs;

    // ---------------- pass 2: weighted SPMM ----------------
    float2 oacc = make_float2(0.f, 0.f);

    if (deg <= SIM_CAP) {
        // Convert sims -> probabilities in place, lane-parallel: one exp per
        // edge instead of one per edge per lane. Same-wave produce/consume,
        // DS ops are in-order within a wave -> no barrier needed.
        for (int i = lane; i < deg; i += 32)
            s_sims[wib][i] = expf(s_sims[wib][i] - runm) * rinv;

        // Pure SPMM stream: uniform 4B p + uniform 4B col + coalesced 256B
        // feature row per edge (float2/lane), accumulated with v_pk_fma_f32.
#pragma unroll 4
        for (int e = start; e < end; ++e) {
            const float pi  = s_sims[wib][e - start];          // uniform LDS read
            const int   ci  = ecol[e];                         // uniform load
            const float2 xv = *(const float2*)(x + (size_t)ci * D_FEAT + 2 * lane);
            oacc.x += pi * xv.x;
            oacc.y += pi * xv.y;
        }
    } else {
        // Fallback (degree > 512): recompute sims with WMMA, broadcast p.
        for (int t = 0; t < ngroups; ++t) {
            const int e  = start + t * 16 + mA;
            const int ec = ecol[(e < end) ? e : start];
            const float nc = norms[ec];
            const float* __restrict__ xb = x + (size_t)ec * D_FEAT + koff;

            v8f acc = {};
#pragma unroll
            for (int c = 0; c < 16; ++c) {
                const float2 a2 = *(const float2*)(xb + 4 * c);
                const float2 b2 = *(const float2*)(xrow + koff + 4 * c);
                v2f a; a.x = a2.x; a.y = a2.y;
                v2f b; b.x = b2.x; b.y = b2.y;
                acc = __builtin_amdgcn_wmma_f32_16x16x4_f32(
                          false, a, false, b, (short)0, acc, false, false);
            }

            float pv[8];
#pragma unroll
            for (int i = 0; i < 8; ++i) {
                const int em = t * 16 + mBase + i;
                const float ncb = __shfl(nc, mBase + i, 32);
                const float s = bscale * acc[i] * __builtin_amdgcn_rcpf(ns * ncb + EPS_F);
                pv[i] = (em < deg) ? (expf(s - runm) * rinv) : 0.f;
            }
#pragma unroll
            for (int i = 0; i < 16; ++i) {
                const float pi = (i < 8) ? __shfl(pv[i], 0, 32)
                                         : __shfl(pv[i - 8], 16, 32);
                const int ci = __shfl(ec, i, 32);              // lane i holds col_i
                const float2 xv = *(const float2*)(x + (size_t)ci * D_FEAT + 2 * lane);
                oacc.x += pi * xv.x;
                oacc.y += pi * xv.y;
            }
        }
    }

    *(float2*)(orow + 2 * lane) = oacc;
}

extern "C" void kernel_launch(void* const* d_in, const int* in_sizes, int n_in,
                              void* d_out, int out_size, void* d_ws, size_t ws_size,
                              hipStream_t stream) {
    const float* x    = (const float*)d_in[0];
    const float* beta = (const float*)d_in[1];
    const int*   erow = (const int*)d_in[2];
    const int*   ecol = (const int*)d_in[3];
    const int n_edges = in_sizes[2];

    float* norms = (float*)d_ws;            // N_NODES floats (400 KB) scratch
    float* out   = (float*)d_out;

    const int threads = 256;                // 8 wave32 waves per block
    const int blocks  = (N_NODES * 32 + threads - 1) / threads;  // 1 wave/row

    node_norm_kernel<<<blocks, threads, 0, stream>>>(x, norms);
    agnn_row_kernel<<<blocks, threads, 0, stream>>>(x, beta, erow, ecol,
                                                    norms, out, n_edges);
}